// HierarchicalCircuit_27144193310735
// MI455X (gfx1250) — compile-verified
//
#include <hip/hip_runtime.h>
#include <math.h>

// ---------------------------------------------------------------------------
// HierarchicalCircuit forward for MI455X (gfx1250, wave32).
// - All GEMMs on the matrix pipeline via V_WMMA_F32_16X16X4_F32 (fp32 in/out,
//   matching the fp32 reference: discrete argmax routing + 32-step recurrence
//   make reduced precision unsafe).
// - incoming@upd_w1[256:]+b1 is loop-invariant across the 32 steps -> hoisted.
// - All LDS tile staging uses GLOBAL_LOAD_ASYNC_TO_LDS_B128 (ASYNCcnt DMA,
//   no VGPR round trip); S writeback uses GLOBAL_STORE_ASYNC_FROM_LDS_B128.
// ---------------------------------------------------------------------------

#define BB   64
#define LL   2048
#define DD   256
#define NN   256      // G*K
#define GG   16
#define KK   16
#define NSTEPS 32

typedef __attribute__((ext_vector_type(2))) float v2f;
typedef __attribute__((ext_vector_type(8))) float v8f;

// (neg_a, A, neg_b, B, c_mod, C, reuse_a, reuse_b)
#define WMMA_F32(a, b, c) \
  __builtin_amdgcn_wmma_f32_16x16x4_f32(false, (a), false, (b), (short)0, (c), false, false)

__device__ __forceinline__ float gelu_exact(float x) {
  return 0.5f * x * (1.0f + erff(x * 0.70710678118654752f));
}

// gfx1250 async DMA: memory -> LDS, 16 bytes per lane, tracked by ASYNCcnt.
// Low 32 bits of a generic LDS pointer are the wave-relative LDS byte offset
// (aperture tag lives in addr[63:32]), so truncation yields the VDST operand.
__device__ __forceinline__ void async_ld16(void* lds_dst, const void* gsrc) {
  unsigned       d = (unsigned)(unsigned long long)lds_dst;
  unsigned long long s = (unsigned long long)gsrc;
  asm volatile("global_load_async_to_lds_b128 %0, %1, off"
               :: "v"(d), "v"(s) : "memory");
}
// LDS -> memory, 16 bytes per lane.
__device__ __forceinline__ void async_st16(const void* gdst, const void* lds_src) {
  unsigned       s = (unsigned)(unsigned long long)lds_src;
  unsigned long long d = (unsigned long long)gdst;
  asm volatile("global_store_async_from_lds_b128 %0, %1, off"
               :: "v"(d), "v"(s) : "memory");
}
__device__ __forceinline__ void async_wait0() {
  asm volatile("s_wait_asynccnt 0x0" ::: "memory");
}

// ---------------------------------------------------------------------------
__global__ void zero_kernel(float* __restrict__ p, long n) {
  long i = (long)blockIdx.x * blockDim.x + threadIdx.x;
  long stride = (long)gridDim.x * blockDim.x;
  for (; i < n; i += stride) p[i] = 0.0f;
}

__global__ void sinit_kernel(float* __restrict__ S, const float* __restrict__ S_init, long n) {
  long i = (long)blockIdx.x * blockDim.x + threadIdx.x;
  long stride = (long)gridDim.x * blockDim.x;
  for (; i < n; i += stride) S[i] = S_init[i & (NN * DD - 1)];  // broadcast over batch
}

// ---------------------------------------------------------------------------
// Query MLP: q[b] = MLP2(concat(embed[qsrc], embed[qtgt])), tiny (64 rows).
__global__ __launch_bounds__(256) void query_kernel(
    const int* __restrict__ qsrc, const int* __restrict__ qtgt,
    const float* __restrict__ node_embed,
    const float* __restrict__ qw1, const float* __restrict__ qb1,
    const float* __restrict__ qw2, const float* __restrict__ qb2,
    float* __restrict__ qbuf) {
  __shared__ __align__(16) float x[512];
  __shared__ float h[256];
  int b = blockIdx.x, t = threadIdx.x;
  int s = qsrc[b]; s = s < 0 ? 0 : (s > 49 ? 49 : s);
  int g = qtgt[b]; g = g < 0 ? 0 : (g > 49 ? 49 : g);
  x[t]       = node_embed[s * DD + t];
  x[256 + t] = node_embed[g * DD + t];
  __syncthreads();
  float acc = qb1[t];
  for (int k = 0; k < 512; ++k) acc += x[k] * qw1[k * 256 + t];
  h[t] = gelu_exact(acc);
  __syncthreads();
  float acc2 = qb2[t];
  for (int k = 0; k < 256; ++k) acc2 += h[k] * qw2[k * 256 + t];
  qbuf[b * DD + t] = acc2;
}

// ---------------------------------------------------------------------------
// Token kernel: one block = 16 tokens.  Edge MLP -> xe, then msg MLP, gate
// MLP, group/slot argmax routing, and atomic scatter of gated messages into
// incoming[b][node].  Query rows (xe = q[b]) are handled by 4 extra blocks.
// 4 waves per block; each wave owns 16-column strips of each WMMA output.
__global__ __launch_bounds__(128) void token_route_kernel(
    const int* __restrict__ src, const int* __restrict__ rel, const int* __restrict__ tgt,
    const float* __restrict__ node_embed, const float* __restrict__ rel_embed,
    const float* __restrict__ edge_w1, const float* __restrict__ edge_b1,
    const float* __restrict__ edge_w2, const float* __restrict__ edge_b2,
    const float* __restrict__ gproj, const float* __restrict__ sproj,
    const float* __restrict__ msg_w1, const float* __restrict__ msg_b1,
    const float* __restrict__ msg_w2, const float* __restrict__ msg_b2,
    const float* __restrict__ gate_w1, const float* __restrict__ gate_b1,
    const float* __restrict__ gate_w2, const float* __restrict__ gate_b2,
    const float* __restrict__ qbuf, float* __restrict__ incoming) {
  __shared__ __align__(16) float xbuf[16 * 768];   // staged x; reused for h1/mh/gh
  __shared__ float gtile[16 * 16];
  __shared__ float stile[16 * 16];
  __shared__ float gatev[16];
  __shared__ int   nodeidx[16];
  __shared__ int   bidx[16];
  __shared__ int   sidx[16], ridx[16], tidx[16];
  __shared__ float red[16 * 8];

  const int tid = threadIdx.x;
  const int lane = tid & 31;
  const int w = tid >> 5;
  const int lo = lane & 15;
  const int hi = lane >> 4;
  const int tileid = blockIdx.x;
  float* xe = xbuf + 8192;  // 16 x 256 region (floats 8192..12287)

  if (tileid < (BB * LL) / 16) {
    // ---------------- edge tiles ----------------
    const int b  = tileid >> 7;           // L/16 = 128 tiles per batch
    const int l0 = (tileid & 127) << 4;
    if (tid < 16) {
      int l = l0 + tid;
      int a = src[b * LL + l]; sidx[tid] = a < 0 ? 0 : (a > 49 ? 49 : a);
      int r = rel[b * LL + l]; ridx[tid] = r < 0 ? 0 : (r > 1 ? 1 : r);
      int c = tgt[b * LL + l]; tidx[tid] = c < 0 ? 0 : (c > 49 ? 49 : c);
      bidx[tid] = b;
    }
    __syncthreads();
    // async-DMA stage x = concat(se, re, te): 16 x 768 in 16B quanta
    for (int e = tid; e < 16 * 192; e += 128) {
      int r = e / 192, c = (e - r * 192) * 4;
      const float* sp;
      if (c < 256)      sp = &node_embed[sidx[r] * DD + c];
      else if (c < 512) sp = &rel_embed[ridx[r] * DD + (c - 256)];
      else              sp = &node_embed[tidx[r] * DD + (c - 512)];
      async_ld16(&xbuf[r * 768 + c], sp);
    }
    async_wait0();
    __syncthreads();
    // GEMM: x(16x768) @ edge_w1(768x256) -> h1, gelu
    v8f acc[4] = {};
    for (int k0 = 0; k0 < 768; k0 += 4) {
      v2f a;
      a.x = xbuf[lo * 768 + k0 + 2 * hi];
      a.y = xbuf[lo * 768 + k0 + 2 * hi + 1];
#pragma unroll
      for (int t = 0; t < 4; ++t) {
        int n0 = w * 64 + t * 16;
        v2f bv;
        bv.x = edge_w1[(k0 + 2 * hi) * 256 + n0 + lo];
        bv.y = edge_w1[(k0 + 2 * hi + 1) * 256 + n0 + lo];
        acc[t] = WMMA_F32(a, bv, acc[t]);
      }
    }
    __syncthreads();  // all reads of x done; reuse xbuf[0..4096) for h1
#pragma unroll
    for (int t = 0; t < 4; ++t) {
      int n = w * 64 + t * 16 + lo;
#pragma unroll
      for (int v = 0; v < 8; ++v)
        xbuf[(v + 8 * hi) * 256 + n] = gelu_exact(acc[t][v] + edge_b1[n]);
    }
    __syncthreads();
    // GEMM: h1(16x256) @ edge_w2(256x256) + b2 -> xe
    v8f acc2[4] = {};
    for (int k0 = 0; k0 < 256; k0 += 4) {
      v2f a;
      a.x = xbuf[lo * 256 + k0 + 2 * hi];
      a.y = xbuf[lo * 256 + k0 + 2 * hi + 1];
#pragma unroll
      for (int t = 0; t < 4; ++t) {
        int n0 = w * 64 + t * 16;
        v2f bv;
        bv.x = edge_w2[(k0 + 2 * hi) * 256 + n0 + lo];
        bv.y = edge_w2[(k0 + 2 * hi + 1) * 256 + n0 + lo];
        acc2[t] = WMMA_F32(a, bv, acc2[t]);
      }
    }
#pragma unroll
    for (int t = 0; t < 4; ++t) {
      int n = w * 64 + t * 16 + lo;
#pragma unroll
      for (int v = 0; v < 8; ++v)
        xe[(v + 8 * hi) * 256 + n] = acc2[t][v] + edge_b2[n];
    }
    __syncthreads();
  } else {
    // ---------------- query tiles (xe = q[b]) ----------------
    const int qt = tileid - (BB * LL) / 16;  // 0..3
    const int b0 = qt * 16;
    if (tid < 16) bidx[tid] = b0 + tid;
    for (int e = tid; e < 16 * 64; e += 128) {  // 16 rows x 64 vec4
      int r = e >> 6, c = (e & 63) * 4;
      async_ld16(&xe[r * 256 + c], &qbuf[(b0 + r) * DD + c]);
    }
    async_wait0();
    __syncthreads();
  }

  // ---- msg hidden: xe @ msg_w1(256x512) + b, gelu -> mh in xbuf[0..8192)
  {
    v8f am[8] = {};
    for (int k0 = 0; k0 < 256; k0 += 4) {
      v2f a;
      a.x = xe[lo * 256 + k0 + 2 * hi];
      a.y = xe[lo * 256 + k0 + 2 * hi + 1];
#pragma unroll
      for (int t = 0; t < 8; ++t) {
        int n0 = w * 128 + t * 16;
        v2f bv;
        bv.x = msg_w1[(k0 + 2 * hi) * 512 + n0 + lo];
        bv.y = msg_w1[(k0 + 2 * hi + 1) * 512 + n0 + lo];
        am[t] = WMMA_F32(a, bv, am[t]);
      }
    }
#pragma unroll
    for (int t = 0; t < 8; ++t) {
      int n = w * 128 + t * 16 + lo;
#pragma unroll
      for (int v = 0; v < 8; ++v)
        xbuf[(v + 8 * hi) * 512 + n] = gelu_exact(am[t][v] + msg_b1[n]);
    }
    __syncthreads();
  }
  // ---- msg out: mh(16x512) @ msg_w2(512x256) + b -> kept in registers
  v8f msgacc[4] = {};
  for (int k0 = 0; k0 < 512; k0 += 4) {
    v2f a;
    a.x = xbuf[lo * 512 + k0 + 2 * hi];
    a.y = xbuf[lo * 512 + k0 + 2 * hi + 1];
#pragma unroll
    for (int t = 0; t < 4; ++t) {
      int n0 = w * 64 + t * 16;
      v2f bv;
      bv.x = msg_w2[(k0 + 2 * hi) * 256 + n0 + lo];
      bv.y = msg_w2[(k0 + 2 * hi + 1) * 256 + n0 + lo];
      msgacc[t] = WMMA_F32(a, bv, msgacc[t]);
    }
  }
#pragma unroll
  for (int t = 0; t < 4; ++t) {
    int n = w * 64 + t * 16 + lo;
#pragma unroll
    for (int v = 0; v < 8; ++v) msgacc[t][v] += msg_b2[n];
  }
  __syncthreads();  // all mh reads done; reuse xbuf[0..4096) for gate hidden
  // ---- gate hidden: xe @ gate_w1(256x256) + b, gelu -> gh in xbuf[0..4096)
  {
    v8f ag[4] = {};
    for (int k0 = 0; k0 < 256; k0 += 4) {
      v2f a;
      a.x = xe[lo * 256 + k0 + 2 * hi];
      a.y = xe[lo * 256 + k0 + 2 * hi + 1];
#pragma unroll
      for (int t = 0; t < 4; ++t) {
        int n0 = w * 64 + t * 16;
        v2f bv;
        bv.x = gate_w1[(k0 + 2 * hi) * 256 + n0 + lo];
        bv.y = gate_w1[(k0 + 2 * hi + 1) * 256 + n0 + lo];
        ag[t] = WMMA_F32(a, bv, ag[t]);
      }
    }
#pragma unroll
    for (int t = 0; t < 4; ++t) {
      int n = w * 64 + t * 16 + lo;
#pragma unroll
      for (int v = 0; v < 8; ++v)
        xbuf[(v + 8 * hi) * 256 + n] = gelu_exact(ag[t][v] + gate_b1[n]);
    }
  }
  // ---- routing logits: xe @ gproj / sproj (256x16), one tile each
  if (w < 2) {
    const float* P = (w == 0) ? gproj : sproj;
    v8f ap = {};
    for (int k0 = 0; k0 < 256; k0 += 4) {
      v2f a;
      a.x = xe[lo * 256 + k0 + 2 * hi];
      a.y = xe[lo * 256 + k0 + 2 * hi + 1];
      v2f bv;
      bv.x = P[(k0 + 2 * hi) * 16 + lo];
      bv.y = P[(k0 + 2 * hi + 1) * 16 + lo];
      ap = WMMA_F32(a, bv, ap);
    }
    float* dst = (w == 0) ? gtile : stile;
#pragma unroll
    for (int v = 0; v < 8; ++v) dst[(v + 8 * hi) * 16 + lo] = ap[v];
  }
  __syncthreads();
  // gate scalar per row: dot(gh[row], gate_w2), 8 threads per row
  {
    int r = tid >> 3, j = tid & 7;
    float s = 0.0f;
    for (int c = j * 32; c < j * 32 + 32; ++c) s += xbuf[r * 256 + c] * gate_w2[c];
    red[r * 8 + j] = s;
  }
  __syncthreads();
  if (tid < 16) {
    float s = 0.0f;
#pragma unroll
    for (int j = 0; j < 8; ++j) s += red[tid * 8 + j];
    s += gate_b2[0];
    gatev[tid] = 1.0f / (1.0f + expf(-s));
    // argmax (first max, matching jnp.argmax)
    int gi = 0; float gm = gtile[tid * 16];
    for (int c = 1; c < GG; ++c) { float v = gtile[tid * 16 + c]; if (v > gm) { gm = v; gi = c; } }
    int si = 0; float sm = stile[tid * 16];
    for (int c = 1; c < KK; ++c) { float v = stile[tid * 16 + c]; if (v > sm) { sm = v; si = c; } }
    nodeidx[tid] = gi * KK + si;
  }
  __syncthreads();
  // scatter gated messages into incoming[b][node]
#pragma unroll
  for (int t = 0; t < 4; ++t) {
    int n = w * 64 + t * 16 + lo;
#pragma unroll
    for (int v = 0; v < 8; ++v) {
      int m = v + 8 * hi;
      float val = msgacc[t][v] * gatev[m];
      atomicAdd(&incoming[((long)bidx[m] * NN + nodeidx[m]) * DD + n], val);
    }
  }
}

// ---------------------------------------------------------------------------
// inc_h = incoming @ upd_w1[256:512,:] + upd_b1   (loop-invariant hoist)
__global__ __launch_bounds__(128) void inch_kernel(
    const float* __restrict__ incoming, const float* __restrict__ upd_w1,
    const float* __restrict__ upd_b1, float* __restrict__ inc_h) {
  __shared__ __align__(16) float abuf[16 * 256];
  int tid = threadIdx.x, lane = tid & 31, w = tid >> 5, lo = lane & 15, hi = lane >> 4;
  long r0 = (long)blockIdx.x * 16;
  for (int e = tid; e < 16 * 64; e += 128)
    async_ld16(&abuf[e * 4], &incoming[r0 * 256 + e * 4]);
  async_wait0();
  __syncthreads();
  v8f acc[8] = {};
  for (int k0 = 0; k0 < 256; k0 += 4) {
    v2f a;
    a.x = abuf[lo * 256 + k0 + 2 * hi];
    a.y = abuf[lo * 256 + k0 + 2 * hi + 1];
#pragma unroll
    for (int t = 0; t < 8; ++t) {
      int n0 = w * 128 + t * 16;
      v2f bv;
      bv.x = upd_w1[(256 + k0 + 2 * hi) * 512 + n0 + lo];
      bv.y = upd_w1[(256 + k0 + 2 * hi + 1) * 512 + n0 + lo];
      acc[t] = WMMA_F32(a, bv, acc[t]);
    }
  }
#pragma unroll
  for (int t = 0; t < 8; ++t) {
    int n = w * 128 + t * 16 + lo;
#pragma unroll
    for (int v = 0; v < 8; ++v)
      inc_h[(r0 + v + 8 * hi) * 512 + n] = acc[t][v] + upd_b1[n];
  }
}

// ---------------------------------------------------------------------------
// One recurrent step, fully fused per 16-row tile:
//   h = gelu(S@W1a + inc_h); u = h@W2 + b2; S = LayerNorm(S + u)
__global__ __launch_bounds__(128) void step_kernel(
    float* __restrict__ S, const float* __restrict__ inc_h,
    const float* __restrict__ upd_w1, const float* __restrict__ upd_w2,
    const float* __restrict__ upd_b2, const float* __restrict__ ln_w,
    const float* __restrict__ ln_b) {
  __shared__ __align__(16) float sbuf[16 * 256];
  __shared__ __align__(16) float hbuf[16 * 512];
  __shared__ float red1[128], red2[128];
  __shared__ float mu_s[16], rs_s[16];
  int tid = threadIdx.x, lane = tid & 31, w = tid >> 5, lo = lane & 15, hi = lane >> 4;
  long r0 = (long)blockIdx.x * 16;
  // async-DMA stage: sbuf <- S rows (16KB), hbuf <- inc_h rows (32KB)
  for (int e = tid; e < 16 * 64; e += 128)
    async_ld16(&sbuf[e * 4], &S[r0 * 256 + e * 4]);
  for (int e = tid; e < 16 * 128; e += 128)
    async_ld16(&hbuf[e * 4], &inc_h[r0 * 512 + e * 4]);
  async_wait0();
  __syncthreads();
  // GEMM1 seeded from staged inc_h (bias folded in); each thread seeds exactly
  // the hbuf elements it will later overwrite, so no extra barrier is needed.
  v8f acc[8];
#pragma unroll
  for (int t = 0; t < 8; ++t) {
    int n = w * 128 + t * 16 + lo;
#pragma unroll
    for (int v = 0; v < 8; ++v) acc[t][v] = hbuf[(v + 8 * hi) * 512 + n];
  }
  for (int k0 = 0; k0 < 256; k0 += 4) {
    v2f a;
    a.x = sbuf[lo * 256 + k0 + 2 * hi];
    a.y = sbuf[lo * 256 + k0 + 2 * hi + 1];
#pragma unroll
    for (int t = 0; t < 8; ++t) {
      int n0 = w * 128 + t * 16;
      v2f bv;
      bv.x = upd_w1[(k0 + 2 * hi) * 512 + n0 + lo];
      bv.y = upd_w1[(k0 + 2 * hi + 1) * 512 + n0 + lo];
      acc[t] = WMMA_F32(a, bv, acc[t]);
    }
  }
#pragma unroll
  for (int t = 0; t < 8; ++t) {
    int n = w * 128 + t * 16 + lo;
#pragma unroll
    for (int v = 0; v < 8; ++v) hbuf[(v + 8 * hi) * 512 + n] = gelu_exact(acc[t][v]);
  }
  __syncthreads();
  // GEMM2: h(16x512) @ upd_w2(512x256)
  v8f acc2[4] = {};
  for (int k0 = 0; k0 < 512; k0 += 4) {
    v2f a;
    a.x = hbuf[lo * 512 + k0 + 2 * hi];
    a.y = hbuf[lo * 512 + k0 + 2 * hi + 1];
#pragma unroll
    for (int t = 0; t < 4; ++t) {
      int n0 = w * 64 + t * 16;
      v2f bv;
      bv.x = upd_w2[(k0 + 2 * hi) * 256 + n0 + lo];
      bv.y = upd_w2[(k0 + 2 * hi + 1) * 256 + n0 + lo];
      acc2[t] = WMMA_F32(a, bv, acc2[t]);
    }
  }
  // residual update in-place (each wave touches only its own columns)
#pragma unroll
  for (int t = 0; t < 4; ++t) {
    int n = w * 64 + t * 16 + lo;
#pragma unroll
    for (int v = 0; v < 8; ++v) {
      int m = v + 8 * hi;
      sbuf[m * 256 + n] += acc2[t][v] + upd_b2[n];
    }
  }
  __syncthreads();
  // LayerNorm over D=256, 8 threads per row
  {
    int r = tid >> 3, j = tid & 7;
    float s = 0.0f, ss = 0.0f;
    for (int c = j * 32; c < j * 32 + 32; ++c) {
      float v = sbuf[r * 256 + c];
      s += v; ss += v * v;
    }
    red1[tid] = s; red2[tid] = ss;
  }
  __syncthreads();
  if (tid < 16) {
    float s = 0.0f, ss = 0.0f;
#pragma unroll
    for (int j = 0; j < 8; ++j) { s += red1[tid * 8 + j]; ss += red2[tid * 8 + j]; }
    float mu = s * (1.0f / 256.0f);
    float var = ss * (1.0f / 256.0f) - mu * mu;
    mu_s[tid] = mu;
    rs_s[tid] = 1.0f / sqrtf(var + 1e-5f);
  }
  __syncthreads();
  {
    int r = tid >> 3, j = tid & 7;
    float mu = mu_s[r], rs = rs_s[r];
    for (int c = j * 32; c < j * 32 + 32; ++c)
      sbuf[r * 256 + c] = (sbuf[r * 256 + c] - mu) * rs * ln_w[c] + ln_b[c];
  }
  __syncthreads();
  // async-DMA writeback: S rows <- sbuf (LDS -> memory, 16B per transfer)
  for (int e = tid; e < 16 * 64; e += 128)
    async_st16(&S[r0 * 256 + e * 4], &sbuf[e * 4]);
  async_wait0();
}

// ---------------------------------------------------------------------------
// Head: logits = MLP2(concat(S[b,qsrc], S[b,qtgt])), 64 rows -> (64,2)
__global__ __launch_bounds__(256) void head_kernel(
    const int* __restrict__ qsrc, const int* __restrict__ qtgt,
    const float* __restrict__ S,
    const float* __restrict__ hw1, const float* __restrict__ hb1,
    const float* __restrict__ hw2, const float* __restrict__ hb2,
    float* __restrict__ out) {
  __shared__ __align__(16) float x[512];
  __shared__ float h[256];
  __shared__ float red[256 * 2];
  int b = blockIdx.x, t = threadIdx.x;
  int s = qsrc[b]; s = s < 0 ? 0 : (s > NN - 1 ? NN - 1 : s);
  int g = qtgt[b]; g = g < 0 ? 0 : (g > NN - 1 ? NN - 1 : g);
  x[t]       = S[((long)b * NN + s) * DD + t];
  x[256 + t] = S[((long)b * NN + g) * DD + t];
  __syncthreads();
  float acc = hb1[t];
  for (int k = 0; k < 512; ++k) acc += x[k] * hw1[k * 256 + t];
  h[t] = gelu_exact(acc);
  __syncthreads();
  red[t * 2]     = h[t] * hw2[t * 2];
  red[t * 2 + 1] = h[t] * hw2[t * 2 + 1];
  __syncthreads();
  if (t < 2) {
    float sum = hb2[t];
    for (int k = 0; k < 256; ++k) sum += red[k * 2 + t];
    out[b * 2 + t] = sum;
  }
}

// ---------------------------------------------------------------------------
extern "C" void kernel_launch(void* const* d_in, const int* in_sizes, int n_in,
                              void* d_out, int out_size, void* d_ws, size_t ws_size,
                              hipStream_t stream) {
  const int*   src        = (const int*)d_in[0];
  const int*   rel        = (const int*)d_in[1];
  const int*   tgt        = (const int*)d_in[2];
  /* d_in[3] = mask, unused by the forward */
  const int*   qsrc       = (const int*)d_in[4];
  const int*   qtgt       = (const int*)d_in[5];
  const float* node_embed = (const float*)d_in[6];
  const float* rel_embed  = (const float*)d_in[7];
  const float* edge_w1    = (const float*)d_in[8];
  const float* edge_b1    = (const float*)d_in[9];
  const float* edge_w2    = (const float*)d_in[10];
  const float* edge_b2    = (const float*)d_in[11];
  const float* query_w1   = (const float*)d_in[12];
  const float* query_b1   = (const float*)d_in[13];
  const float* query_w2   = (const float*)d_in[14];
  const float* query_b2   = (const float*)d_in[15];
  const float* gproj      = (const float*)d_in[16];
  const float* sproj      = (const float*)d_in[17];
  const float* msg_w1     = (const float*)d_in[18];
  const float* msg_b1     = (const float*)d_in[19];
  const float* msg_w2     = (const float*)d_in[20];
  const float* msg_b2     = (const float*)d_in[21];
  const float* gate_w1    = (const float*)d_in[22];
  const float* gate_b1    = (const float*)d_in[23];
  const float* gate_w2    = (const float*)d_in[24];
  const float* gate_b2    = (const float*)d_in[25];
  const float* upd_w1     = (const float*)d_in[26];
  const float* upd_b1     = (const float*)d_in[27];
  const float* upd_w2     = (const float*)d_in[28];
  const float* upd_b2     = (const float*)d_in[29];
  const float* ln_w       = (const float*)d_in[30];
  const float* ln_b       = (const float*)d_in[31];
  const float* S_init     = (const float*)d_in[32];
  const float* head_w1    = (const float*)d_in[33];
  const float* head_b1    = (const float*)d_in[34];
  const float* head_w2    = (const float*)d_in[35];
  const float* head_b2    = (const float*)d_in[36];

  // Workspace layout (floats): q | incoming | inc_h | S  == 64.1 MB total
  float* ws       = (float*)d_ws;
  float* qbuf     = ws;                                 // 64*256
  float* incoming = qbuf + (long)BB * DD;               // 64*256*256
  float* inch     = incoming + (long)BB * NN * DD;      // 64*256*512
  float* Sbuf     = inch + (long)BB * NN * 2 * DD;      // 64*256*256

  const long incN = (long)BB * NN * DD;

  zero_kernel<<<2048, 256, 0, stream>>>(incoming, incN);
  query_kernel<<<BB, 256, 0, stream>>>(qsrc, qtgt, node_embed,
                                       query_w1, query_b1, query_w2, query_b2, qbuf);
  token_route_kernel<<<(BB * LL) / 16 + BB / 16, 128, 0, stream>>>(
      src, rel, tgt, node_embed, rel_embed,
      edge_w1, edge_b1, edge_w2, edge_b2,
      gproj, sproj, msg_w1, msg_b1, msg_w2, msg_b2,
      gate_w1, gate_b1, gate_w2, gate_b2, qbuf, incoming);
  inch_kernel<<<(BB * NN) / 16, 128, 0, stream>>>(incoming, upd_w1, upd_b1, inch);
  sinit_kernel<<<2048, 256, 0, stream>>>(Sbuf, S_init, incN);
  for (int s = 0; s < NSTEPS; ++s)
    step_kernel<<<(BB * NN) / 16, 128, 0, stream>>>(Sbuf, inch, upd_w1, upd_w2,
                                                    upd_b2, ln_w, ln_b);
  head_kernel<<<BB, 256, 0, stream>>>(qsrc, qtgt, Sbuf,
                                      head_w1, head_b1, head_w2, head_b2,
                                      (float*)d_out);
}